// Het_Agg_66692252172828
// MI455X (gfx1250) — compile-verified
//
#include <hip/hip_runtime.h>
#include <math.h>

// Het_Agg for MI455X (gfx1250): 3x WMMA-f32 GEMM+ReLU, fp32 HW-atomic edge
// scatter, streaming softmax-attention combine, fused WMMA GEMM + L2-norm.

static constexpr int DIM = 256;   // feature dim D

typedef __attribute__((ext_vector_type(2))) float v2f;
typedef __attribute__((ext_vector_type(8))) float v8f;

// ---------------------------------------------------------------------------
// H = relu(X @ W^T + bias).  X:[n,256] W:[256,256] row-major.
// One wave per 16x16 output tile; K-loop of 64 x v_wmma_f32_16x16x4_f32.
// A-frag: row (lane&15), K pair 2*(lane>>4).  B[k][j] = W[j][k] -> contiguous
// float2 load from W row (jbase + lane&15).
// ---------------------------------------------------------------------------
__global__ __launch_bounds__(256) void hetagg_gemm_relu(
    const float* __restrict__ X, const float* __restrict__ W,
    const float* __restrict__ bias, float* __restrict__ H, int nTiles)
{
    const int wave = threadIdx.x >> 5;
    const int lane = threadIdx.x & 31;
    const int tile = blockIdx.x * 8 + wave;
    if (tile >= nTiles) return;                 // wave-uniform: EXEC stays full
    const int ibase = (tile >> 4) * 16;         // 16 column tiles (256/16)
    const int jbase = (tile & 15) * 16;
    const int m  = lane & 15;
    const int kp = lane >> 4;                   // K sub-pair: 0 or 1

    const float* arow = X + (size_t)(ibase + m) * DIM + 2 * kp;
    const float* brow = W + (size_t)(jbase + m) * DIM + 2 * kp;

    v8f c = {};
#pragma unroll 8
    for (int k = 0; k < DIM; k += 4) {
        v2f a = *(const v2f*)(arow + k);
        v2f b = *(const v2f*)(brow + k);
        c = __builtin_amdgcn_wmma_f32_16x16x4_f32(false, a, false, b,
                                                  (short)0, c, false, false);
    }
    const int col = jbase + m;
    const float bb = bias[col];
#pragma unroll
    for (int r = 0; r < 8; ++r) {
        const int row = ibase + r + 8 * kp;     // C/D layout: M = r + 8*(lane>>4)
        float v = c[r] + bb;
        H[(size_t)row * DIM + col] = v > 0.f ? v : 0.f;
    }
}

// ---------------------------------------------------------------------------
// Edge scatter: aggr[src] += H[tgt] * ew[e]; deg[src] += 1.
// One wave per edge, 8 floats per lane, hardware global_atomic_add_f32.
// ---------------------------------------------------------------------------
__global__ __launch_bounds__(256) void hetagg_scatter(
    const float* __restrict__ H, const int* __restrict__ edge,
    const float* __restrict__ ew, float* __restrict__ aggr,
    float* __restrict__ deg, int E)
{
    const int wave = threadIdx.x >> 5;
    const int lane = threadIdx.x & 31;
    const int e = blockIdx.x * 8 + wave;
    if (e >= E) return;                          // wave-uniform
    const int   src = edge[e];
    const int   tgt = edge[E + e];
    const float w   = ew[e];

    const float4* hrow = (const float4*)(H + (size_t)tgt * DIM) + lane * 2;
    const float4 h0 = hrow[0];
    const float4 h1 = hrow[1];
    float* arow = aggr + (size_t)src * DIM + lane * 8;
    unsafeAtomicAdd(arow + 0, h0.x * w);
    unsafeAtomicAdd(arow + 1, h0.y * w);
    unsafeAtomicAdd(arow + 2, h0.z * w);
    unsafeAtomicAdd(arow + 3, h0.w * w);
    unsafeAtomicAdd(arow + 4, h1.x * w);
    unsafeAtomicAdd(arow + 5, h1.y * w);
    unsafeAtomicAdd(arow + 6, h1.z * w);
    unsafeAtomicAdd(arow + 7, h1.w * w);
    if (lane == 0) unsafeAtomicAdd(&deg[src], 1.0f);
}

// ---------------------------------------------------------------------------
// Per-channel attention accumulation (streaming softmax):
//   an  = aggr[i] / max(deg[i],1)
//   s   = exp(leaky_relu(an . u1 + x_node . u2, 0.01))
//   num[i] += s * an;  den[i] += s
// One wave per node; wave reduction via __shfl_xor (wave32).
// ---------------------------------------------------------------------------
__global__ __launch_bounds__(256) void hetagg_score(
    const float* __restrict__ aggr, const float* __restrict__ deg,
    const float* __restrict__ xnode, const float* __restrict__ u,
    float* __restrict__ num, float* __restrict__ den, int n)
{
    const int wave = threadIdx.x >> 5;
    const int lane = threadIdx.x & 31;
    const int i = blockIdx.x * 8 + wave;
    if (i >= n) return;                          // wave-uniform
    const float dinv = 1.0f / fmaxf(deg[i], 1.0f);
    const int j0 = lane * 8;
    const float* ar = aggr  + (size_t)i * DIM + j0;
    const float* xr = xnode + (size_t)i * DIM + j0;

    float an[8];
    float part = 0.f;
#pragma unroll
    for (int t = 0; t < 8; ++t) {
        an[t] = ar[t] * dinv;
        part += an[t] * u[j0 + t] + xr[t] * u[DIM + j0 + t];
    }
#pragma unroll
    for (int off = 16; off > 0; off >>= 1)
        part += __shfl_xor(part, off, 32);

    const float z = part > 0.f ? part : 0.01f * part;
    const float s = expf(z);

    float* nr = num + (size_t)i * DIM + j0;
#pragma unroll
    for (int t = 0; t < 8; ++t) nr[t] += s * an[t];
    if (lane == 0) den[i] += s;
}

// ---------------------------------------------------------------------------
// Final: cat = [x_node | num/den] (N x 512), h = relu(cat @ Wlin^T + blin),
// out = h / max(||h||_2, 1e-12).  One block per 16 rows; 8 waves x 2 col
// tiles = full 256-col band, K=512 split as 256 (x_node) + 256 (num*invden).
// Row norms via LDS.
// ---------------------------------------------------------------------------
__global__ __launch_bounds__(256) void hetagg_final(
    const float* __restrict__ xnode, const float* __restrict__ num,
    const float* __restrict__ den, const float* __restrict__ Wl,  // [256,512]
    const float* __restrict__ bl, float* __restrict__ out, int n)
{
    __shared__ float sh[16 * DIM];
    __shared__ float sinv[16];
    const int wave = threadIdx.x >> 5;
    const int lane = threadIdx.x & 31;
    const int ibase = blockIdx.x * 16;
    const int m  = lane & 15;
    const int kp = lane >> 4;

    const float invden = 1.0f / den[ibase + m];
    const float* xrow = xnode + (size_t)(ibase + m) * DIM + 2 * kp;
    const float* crow = num   + (size_t)(ibase + m) * DIM + 2 * kp;

    for (int t = 0; t < 2; ++t) {
        const int jbase = (wave * 2 + t) * 16;
        const float* wrow = Wl + (size_t)(jbase + m) * (2 * DIM) + 2 * kp;
        v8f c = {};
#pragma unroll 8
        for (int k = 0; k < DIM; k += 4) {       // K = 0..255 : x_node part
            v2f a = *(const v2f*)(xrow + k);
            v2f b = *(const v2f*)(wrow + k);
            c = __builtin_amdgcn_wmma_f32_16x16x4_f32(false, a, false, b,
                                                      (short)0, c, false, false);
        }
#pragma unroll 8
        for (int k = 0; k < DIM; k += 4) {       // K = 256..511 : combined part
            v2f a = *(const v2f*)(crow + k);
            a = a * invden;
            v2f b = *(const v2f*)(wrow + DIM + k);
            c = __builtin_amdgcn_wmma_f32_16x16x4_f32(false, a, false, b,
                                                      (short)0, c, false, false);
        }
        const int col = jbase + m;
        const float bb = bl[col];
#pragma unroll
        for (int r = 0; r < 8; ++r) {
            float v = c[r] + bb;
            sh[(r + 8 * kp) * DIM + col] = v > 0.f ? v : 0.f;
        }
    }
    __syncthreads();
    if (threadIdx.x < 16) {
        const float* row = sh + threadIdx.x * DIM;
        float sum = 0.f;
        for (int j = 0; j < DIM; ++j) sum += row[j] * row[j];
        sinv[threadIdx.x] = 1.0f / fmaxf(sqrtf(sum), 1e-12f);
    }
    __syncthreads();
#pragma unroll
    for (int r = 0; r < 16; ++r)
        out[(size_t)(ibase + r) * DIM + threadIdx.x] =
            sh[r * DIM + threadIdx.x] * sinv[r];
}

// ---------------------------------------------------------------------------
// Host side: graph-capture-safe (kernels + hipMemsetAsync only).
// Workspace layout (floats): h[N*D] | aggr[N*D] | num[N*D] | deg[N] | den[N]
// ---------------------------------------------------------------------------
extern "C" void kernel_launch(void* const* d_in, const int* in_sizes, int n_in,
                              void* d_out, int out_size, void* d_ws, size_t ws_size,
                              hipStream_t stream)
{
    const float* x_a    = (const float*)d_in[0];
    const float* x_p    = (const float*)d_in[1];
    const float* x_v    = (const float*)d_in[2];
    const int*   edge_a = (const int*)  d_in[3];
    const int*   edge_p = (const int*)  d_in[4];
    const int*   edge_v = (const int*)  d_in[5];
    const float* ew_a   = (const float*)d_in[6];
    const float* ew_p   = (const float*)d_in[7];
    const float* ew_v   = (const float*)d_in[8];
    const float* x_node = (const float*)d_in[9];
    // d_in[10] = num_node scalar on device; derive N from in_sizes instead.
    const float* W_a    = (const float*)d_in[11];
    const float* b_a    = (const float*)d_in[12];
    const float* W_p    = (const float*)d_in[13];
    const float* b_p    = (const float*)d_in[14];
    const float* W_v    = (const float*)d_in[15];
    const float* b_v    = (const float*)d_in[16];
    const float* u      = (const float*)d_in[17];
    const float* W_lin  = (const float*)d_in[18];
    const float* b_lin  = (const float*)d_in[19];
    float* out = (float*)d_out;

    const int N = in_sizes[0] / DIM;     // 50000
    const int E = in_sizes[6];           // 300000

    float* ws   = (float*)d_ws;
    float* h    = ws;
    float* aggr = ws + (size_t)N * DIM;
    float* num  = ws + 2 * (size_t)N * DIM;
    float* deg  = ws + 3 * (size_t)N * DIM;
    float* den  = deg + N;

    hipMemsetAsync(num, 0, sizeof(float) * (size_t)N * DIM, stream);
    hipMemsetAsync(den, 0, sizeof(float) * (size_t)N, stream);

    const float* Xs[3]  = {x_a, x_p, x_v};
    const float* Ws[3]  = {W_a, W_p, W_v};
    const float* Bs[3]  = {b_a, b_p, b_v};
    const int*   Es[3]  = {edge_a, edge_p, edge_v};
    const float* EWs[3] = {ew_a, ew_p, ew_v};

    const int nTiles     = (N / 16) * (DIM / 16);      // 16x16 output tiles
    const int gemmBlocks = (nTiles + 7) / 8;
    const int scatBlocks = (E + 7) / 8;
    const int scoreBlocks = (N + 7) / 8;

    for (int ch = 0; ch < 3; ++ch) {
        hetagg_gemm_relu<<<gemmBlocks, 256, 0, stream>>>(Xs[ch], Ws[ch], Bs[ch],
                                                         h, nTiles);
        hipMemsetAsync(aggr, 0, sizeof(float) * (size_t)N * DIM, stream);
        hipMemsetAsync(deg,  0, sizeof(float) * (size_t)N, stream);
        hetagg_scatter<<<scatBlocks, 256, 0, stream>>>(h, Es[ch], EWs[ch],
                                                       aggr, deg, E);
        hetagg_score<<<scoreBlocks, 256, 0, stream>>>(aggr, deg, x_node, u,
                                                      num, den, N);
    }
    hetagg_final<<<N / 16, 256, 0, stream>>>(x_node, num, den, W_lin, b_lin,
                                             out, N);
    (void)n_in; (void)out_size; (void)ws_size;
}